// VolumeRendererMip_60696477827100
// MI455X (gfx1250) — compile-verified
//
#include <hip/hip_runtime.h>
#include <cstdint>
#include <cstddef>

// ---------------------------------------------------------------------------
// NeRF-style volume renderer for MI455X (gfx1250).
//   - one ray per wave32, 4 samples per lane
//   - exclusive cumsum via __shfl_up wave scan (wave32-sized)
//   - density/rgbs streamed through LDS with global_load_async_to_lds_b128
//     double buffering (ASYNCcnt), overlapping DRAM with the threefry RNG
// ---------------------------------------------------------------------------

#define S 128
#define WAVES_PER_BLOCK 8
#define ITERS 4
#define RAYS_PER_BLOCK (WAVES_PER_BLOCK * ITERS)   // 32
// LDS slot per ray: 128 density floats (512B) + 384 rgb floats (1536B) = 512 floats
#define SLOT_FLOATS 512

__device__ __forceinline__ uint32_t rotl32(uint32_t x, int r) {
    return (x << r) | (x >> (32 - r));
}

// Exact reproduction of jax.random.normal(jax.random.key(42), (n,), f32) at
// flat index idx (threefry2x32, JAX split-counter layout, erfinv transform).
__device__ __forceinline__ float jax_threefry_normal(uint32_t idx, uint32_t halfN) {
    uint32_t x0, x1;
    const bool lo_half = idx < halfN;
    if (lo_half) { x0 = idx;         x1 = idx + halfN; }
    else         { x0 = idx - halfN; x1 = idx;         }
    const uint32_t k0 = 0u, k1 = 42u;
    const uint32_t k2 = 0x1BD11BDAu ^ k0 ^ k1;
    x0 += k0; x1 += k1;
#define TF_R(r) { x0 += x1; x1 = rotl32(x1, r); x1 ^= x0; }
    TF_R(13) TF_R(15) TF_R(26) TF_R(6)
    x0 += k1; x1 += k2 + 1u;
    TF_R(17) TF_R(29) TF_R(16) TF_R(24)
    x0 += k2; x1 += k0 + 2u;
    TF_R(13) TF_R(15) TF_R(26) TF_R(6)
    x0 += k0; x1 += k1 + 3u;
    TF_R(17) TF_R(29) TF_R(16) TF_R(24)
    x0 += k1; x1 += k2 + 4u;
    TF_R(13) TF_R(15) TF_R(26) TF_R(6)
    x0 += k2; x1 += k0 + 5u;
#undef TF_R
    const uint32_t bits = lo_half ? x0 : x1;
    const float r01 = __uint_as_float((bits >> 9) | 0x3f800000u) - 1.0f;  // [0,1)
    const float lo = -0.99999994f;                                        // nextafter(-1,0)
    const float u = fmaxf(lo, fmaf(r01, 1.0f - lo, lo));
    return 1.4142135623730951f * erfinvf(u);
}

__device__ __forceinline__ float softplusf(float x) {
    return fmaxf(x, 0.0f) + log1pf(expf(-fabsf(x)));
}
__device__ __forceinline__ float sigmoidf(float x) {
    return 1.0f / (1.0f + expf(-x));
}
__device__ __forceinline__ float wave_sum(float v) {
#pragma unroll
    for (int d = 16; d > 0; d >>= 1) v += __shfl_xor(v, d, 32);
    return v;
}

// Issue 4 async B128 copies: 512B density + 1536B rgbs for one ray into the
// wave's LDS slot. Each op: 32 lanes x 16B.  Tracked by ASYNCcnt.
__device__ __forceinline__ void async_issue_ray(const float* __restrict__ dsrc,
                                                const float* __restrict__ rsrc,
                                                uint32_t ldsBase, int lane) {
    uint64_t gd = (uint64_t)(uintptr_t)dsrc + (uint64_t)(lane * 16);
    uint32_t ld = ldsBase + (uint32_t)(lane * 16);
    asm volatile("global_load_async_to_lds_b128 %0, %1, off"
                 :: "v"(ld), "v"(gd) : "memory");
    uint64_t gr = (uint64_t)(uintptr_t)rsrc + (uint64_t)(lane * 16);
    uint32_t lr = ldsBase + 2048u + (uint32_t)(lane * 16);
#pragma unroll
    for (int k = 0; k < 3; ++k) {
        asm volatile("global_load_async_to_lds_b128 %0, %1, off"
                     :: "v"(lr), "v"(gr) : "memory");
        gr += 512u; lr += 512u;
    }
}

__global__ __launch_bounds__(256)
void volume_render_kernel(const float* __restrict__ density,
                          const float* __restrict__ rgbs,
                          const float* __restrict__ t_vals,
                          const float* __restrict__ dirs,
                          float* __restrict__ out,
                          int nRays, uint32_t halfN) {
    __shared__ __align__(16) float lds[2][WAVES_PER_BLOCK][SLOT_FLOATS];  // 32 KB

    const int tid  = threadIdx.x;
    const int lane = tid & 31;
    const int w    = tid >> 5;
    const int rayBase = blockIdx.x * RAYS_PER_BLOCK;

    const uint32_t ldsBase0 = (uint32_t)(uintptr_t)(&lds[0][w][0]);
    const uint32_t ldsBase1 = (uint32_t)(uintptr_t)(&lds[1][w][0]);

    float* __restrict__ outRGB   = out;
    float* __restrict__ outDepth = out + (size_t)nRays * 3;
    float* __restrict__ outAcc   = out + (size_t)nRays * 4;
    float* __restrict__ outW     = out + (size_t)nRays * 5;

    // Prologue: prefetch iteration 0.
    {
        const int ray = rayBase + w;
        async_issue_ray(density + (size_t)ray * S, rgbs + (size_t)ray * S * 3,
                        ldsBase0, lane);
    }

    for (int it = 0; it < ITERS; ++it) {
        const int ray = rayBase + it * WAVES_PER_BLOCK + w;
        const float* slot = (it & 1) ? &lds[1][w][0] : &lds[0][w][0];

        if (it + 1 < ITERS) {
            const int nray = ray + WAVES_PER_BLOCK;
            const uint32_t nbase = ((it + 1) & 1) ? ldsBase1 : ldsBase0;
            async_issue_ray(density + (size_t)nray * S, rgbs + (size_t)nray * S * 3,
                            nbase, lane);
            // 8 outstanding max; in-order completion => <=4 left means the
            // current iteration's 4 copies have landed in LDS.
            asm volatile("s_wait_asynccnt 4" ::: "memory");
        } else {
            asm volatile("s_wait_asynccnt 0" ::: "memory");
        }

        // ---- t values (ray*(S+1) base is only 4B aligned -> scalar loads) ----
        const float* tv = t_vals + (size_t)ray * (S + 1) + lane * 4;
        const float t0 = tv[0], t1 = tv[1], t2 = tv[2], t3 = tv[3], t4 = tv[4];

        const float* dp = dirs + (size_t)ray * 3;
        const float dn = sqrtf(dp[0] * dp[0] + dp[1] * dp[1] + dp[2] * dp[2]);

        // ---- density_delta for this lane's 4 samples ----
        const float4 dv = *(const float4*)(slot + lane * 4);
        const uint32_t idx0 = (uint32_t)ray * (uint32_t)S + (uint32_t)(lane * 4);
        const float dd0 = softplusf(dv.x + jax_threefry_normal(idx0 + 0u, halfN) - 1.0f) * (t1 - t0) * dn;
        const float dd1 = softplusf(dv.y + jax_threefry_normal(idx0 + 1u, halfN) - 1.0f) * (t2 - t1) * dn;
        const float dd2 = softplusf(dv.z + jax_threefry_normal(idx0 + 2u, halfN) - 1.0f) * (t3 - t2) * dn;
        const float dd3 = softplusf(dv.w + jax_threefry_normal(idx0 + 3u, halfN) - 1.0f) * (t4 - t3) * dn;

        const float p1 = dd0, p2 = p1 + dd1, p3 = p2 + dd2;
        const float tot = p3 + dd3;

        // ---- wave32 inclusive scan of lane totals -> exclusive base ----
        float inc = tot;
#pragma unroll
        for (int d = 1; d < 32; d <<= 1) {
            const float nv = __shfl_up(inc, d, 32);
            if (lane >= d) inc += nv;
        }
        const float base = inc - tot;

        // weights: w_j = T_j - T_{j+1}
        const float tr0 = expf(-base);
        const float tr1 = expf(-(base + p1));
        const float tr2 = expf(-(base + p2));
        const float tr3 = expf(-(base + p3));
        const float tr4 = expf(-inc);
        const float w0 = tr0 - tr1, w1 = tr1 - tr2, w2 = tr2 - tr3, w3 = tr3 - tr4;

        // ---- rgb composite ----
        const float* rs = slot + 512 + lane * 12;
        const float4 ra = *(const float4*)(rs);
        const float4 rb = *(const float4*)(rs + 4);
        const float4 rc = *(const float4*)(rs + 8);
        float sr = w0 * sigmoidf(ra.x) + w1 * sigmoidf(ra.w) + w2 * sigmoidf(rb.z) + w3 * sigmoidf(rc.y);
        float sg = w0 * sigmoidf(ra.y) + w1 * sigmoidf(rb.x) + w2 * sigmoidf(rb.w) + w3 * sigmoidf(rc.z);
        float sb = w0 * sigmoidf(ra.z) + w1 * sigmoidf(rb.y) + w2 * sigmoidf(rc.x) + w3 * sigmoidf(rc.w);
        float sw = w0 + w1 + w2 + w3;
        const float tm0 = 0.5f * (t0 + t1), tm1 = 0.5f * (t1 + t2);
        const float tm2 = 0.5f * (t2 + t3), tm3 = 0.5f * (t3 + t4);
        float st = w0 * tm0 + w1 * tm1 + w2 * tm2 + w3 * tm3;

        sr = wave_sum(sr); sg = wave_sum(sg); sb = wave_sum(sb);
        sw = wave_sum(sw); st = wave_sum(st);

        // weights out (coalesced float4 per lane)
        *(float4*)(outW + (size_t)ray * S + lane * 4) = make_float4(w0, w1, w2, w3);

        const float tlast = __shfl(t4, 31, 32);   // t_vals[ray][S]
        if (lane == 0) {
            outRGB[(size_t)ray * 3 + 0] = sr;
            outRGB[(size_t)ray * 3 + 1] = sg;
            outRGB[(size_t)ray * 3 + 2] = sb;
            float depth = st / sw;
            if (isnan(depth)) depth = __builtin_inff();
            depth = fminf(fmaxf(depth, t0), tlast);  // t0 on lane 0 == t_vals[ray][0]
            outDepth[ray] = depth;
            outAcc[ray]   = sw;
        }
    }
}

extern "C" void kernel_launch(void* const* d_in, const int* in_sizes, int n_in,
                              void* d_out, int out_size, void* d_ws, size_t ws_size,
                              hipStream_t stream) {
    (void)n_in; (void)out_size; (void)d_ws; (void)ws_size;
    const float* density = (const float*)d_in[0];
    const float* rgbs    = (const float*)d_in[1];
    const float* t_vals  = (const float*)d_in[2];
    const float* dirs    = (const float*)d_in[3];
    float* out = (float*)d_out;

    const int nRays = in_sizes[3] / 3;                       // dirs is [N,3]
    const uint32_t halfN = (uint32_t)(((uint64_t)nRays * S) / 2);
    const int blocks = nRays / RAYS_PER_BLOCK;               // 65536/32 = 2048

    hipLaunchKernelGGL(volume_render_kernel, dim3(blocks), dim3(256), 0, stream,
                       density, rgbs, t_vals, dirs, out, nRays, halfN);
}